// Sampler_21182778704451
// MI455X (gfx1250) — compile-verified
//
#include <hip/hip_runtime.h>
#include <math.h>
#include <stdint.h>

#define B_  64
#define S_  16
#define D_  2048
#define V_  128000
#define TOPC 64

typedef __attribute__((ext_vector_type(2))) float f32x2;
typedef __attribute__((ext_vector_type(8))) float f32x8;

// ---------- K1: logits = (hs @ emb^T) / temp, via fp32 WMMA (16x16x4) ----------
// One wave per 16-wide vocab tile, covering all 64 batch rows (4 M-tiles).
// hs K-window (64 rows x 16 k, fp32 = 4KB) is staged into LDS with async
// loads (double-buffered), shared by all 8 waves of the block.
__global__ void __launch_bounds__(256)
k_logits_wmma(const float* __restrict__ emb, const float* __restrict__ hs,
              const int* __restrict__ outpos, const float* __restrict__ temps,
              float* __restrict__ logits) {
    constexpr int KW = 16;                // K-window
    constexpr int NW = D_ / KW;           // 128 windows
    __shared__ float hsLds[2][B_ * KW];   // 2 x 4KB

    const int t    = threadIdx.x;
    const int wid  = t >> 5;
    const int lane = t & 31;
    const int nt   = blockIdx.x * 8 + wid;         // vocab tile 0..7999
    const int n0   = nt * 16;
    const int half = lane >> 4;
    const int ln   = lane & 15;
    const int pos  = outpos[0];

    // staging map: thread t -> hs row (t/4), 4-float chunk (t%4)
    const int tr = t >> 2;
    const int tc = (t & 3) * 4;
    const float* __restrict__ hsrc = hs + ((size_t)tr * S_ + pos) * D_ + tc;
    const uint32_t ldsBase = (uint32_t)(uintptr_t)&hsLds[0][0];
    const uint32_t myDst   = ldsBase + (uint32_t)t * 16u;   // 16B per thread per window

    const float* __restrict__ embRow = emb + (size_t)(n0 + ln) * D_;

    f32x8 acc0 = {}, acc1 = {}, acc2 = {}, acc3 = {};

    // prologue: stage window 0 into buffer 0
    {
        uint64_t ga = (uint64_t)(uintptr_t)hsrc;
        asm volatile("global_load_async_to_lds_b128 %0, %1, off"
                     :: "v"(myDst), "v"(ga) : "memory");
    }

    for (int i = 0; i < NW; ++i) {
        const int k0 = i * KW;
        __syncthreads();   // all waves done reading buf[(i+1)&1] (iter i-1)

        if (i + 1 < NW) {
            uint64_t ga = (uint64_t)(uintptr_t)(hsrc + (i + 1) * KW);
            uint32_t dst = ldsBase + (uint32_t)(((i + 1) & 1) * (B_ * KW * 4)) + (uint32_t)t * 16u;
            asm volatile("global_load_async_to_lds_b128 %0, %1, off"
                         :: "v"(dst), "v"(ga) : "memory");
            asm volatile("s_wait_asynccnt 0x1" ::: "memory"); // window i landed (this wave)
        } else {
            asm volatile("s_wait_asynccnt 0x0" ::: "memory");
        }
        __syncthreads();   // every wave has waited -> window i fully in LDS

        __builtin_prefetch(embRow + k0 + 256, 0, 1);          // global_prefetch_b8

        const float* buf = &hsLds[i & 1][0];
        const int ke = 2 * half;                               // lane's k sub-offset
#pragma unroll
        for (int s = 0; s < 4; ++s) {                          // 4 WMMA K-steps of 4
            const int kk = 4 * s + ke;
            f32x2 bfr = *reinterpret_cast<const f32x2*>(embRow + k0 + kk);
            f32x2 a0  = *reinterpret_cast<const f32x2*>(buf + ( 0 + ln) * KW + kk);
            f32x2 a1  = *reinterpret_cast<const f32x2*>(buf + (16 + ln) * KW + kk);
            f32x2 a2  = *reinterpret_cast<const f32x2*>(buf + (32 + ln) * KW + kk);
            f32x2 a3  = *reinterpret_cast<const f32x2*>(buf + (48 + ln) * KW + kk);
            acc0 = __builtin_amdgcn_wmma_f32_16x16x4_f32(false, a0, false, bfr, (short)0, acc0, false, false);
            acc1 = __builtin_amdgcn_wmma_f32_16x16x4_f32(false, a1, false, bfr, (short)0, acc1, false, false);
            acc2 = __builtin_amdgcn_wmma_f32_16x16x4_f32(false, a2, false, bfr, (short)0, acc2, false, false);
            acc3 = __builtin_amdgcn_wmma_f32_16x16x4_f32(false, a3, false, bfr, (short)0, acc3, false, false);
        }
    }

    // C/D layout: lane ln = column n, VGPR r -> row r + 8*half (per M-tile)
    const int n = n0 + ln;
#pragma unroll
    for (int r = 0; r < 8; ++r) {
        int m = r + 8 * half;
        logits[(size_t)(m +  0) * V_ + n] = acc0[r] / temps[m +  0];
        logits[(size_t)(m + 16) * V_ + n] = acc1[r] / temps[m + 16];
        logits[(size_t)(m + 32) * V_ + n] = acc2[r] / temps[m + 32];
        logits[(size_t)(m + 48) * V_ + n] = acc3[r] / temps[m + 48];
    }
}

// ---------- K2: per-row max and sum(exp) ----------
__global__ void __launch_bounds__(1024)
k_rowstats(const float* __restrict__ logits, float* __restrict__ rowMax, float* __restrict__ rowZ) {
    const int b = blockIdx.x, t = threadIdx.x;
    __shared__ float red[1024];
    const float* row = logits + (size_t)b * V_;
    float m = -__builtin_huge_valf();
    for (int v = t; v < V_; v += 1024) m = fmaxf(m, row[v]);
    red[t] = m; __syncthreads();
    for (int s = 512; s > 0; s >>= 1) { if (t < s) red[t] = fmaxf(red[t], red[t + s]); __syncthreads(); }
    const float rm = red[0]; __syncthreads();
    float z = 0.f;
    for (int v = t; v < V_; v += 1024) z += __expf(row[v] - rm);
    red[t] = z; __syncthreads();
    for (int s = 512; s > 0; s >>= 1) { if (t < s) red[t] += red[t + s]; __syncthreads(); }
    if (t == 0) { rowMax[b] = rm; rowZ[b] = red[0]; }
}

// ---------- K3: per-row top-64 (value desc, stable index tie-break) ----------
__global__ void __launch_bounds__(1024)
k_top64(float* __restrict__ logits, float* __restrict__ topv, int* __restrict__ topi) {
    const int b = blockIdx.x, t = threadIdx.x;
    const int CH = V_ / 1024;                       // 125
    __shared__ float lv[1024]; __shared__ int li[1024];
    __shared__ float rv[1024]; __shared__ int ri[1024];
    float* row = logits + (size_t)b * V_;
    const int s0 = t * CH;

    float bv = -__builtin_huge_valf(); int bi = s0;
    for (int i = 0; i < CH; ++i) { float x = row[s0 + i]; if (x > bv) { bv = x; bi = s0 + i; } }
    lv[t] = bv; li[t] = bi;

    for (int r = 0; r < TOPC; ++r) {
        __syncthreads();
        rv[t] = lv[t]; ri[t] = li[t];
        __syncthreads();
        for (int s = 512; s > 0; s >>= 1) {
            if (t < s) {
                float va = rv[t], vb = rv[t + s]; int ia = ri[t], ib = ri[t + s];
                if (vb > va || (vb == va && ib < ia)) { rv[t] = vb; ri[t] = ib; }
            }
            __syncthreads();
        }
        const float wv = rv[0]; const int wi = ri[0];
        if (t == 0) { topv[b * TOPC + r] = wv; topi[b * TOPC + r] = wi; }
        if (t == wi / CH) {                         // owner knocks out its winner, rescans
            row[wi] = -__builtin_huge_valf();
            float nv = -__builtin_huge_valf(); int ni = s0;
            for (int i = 0; i < CH; ++i) { float x = row[s0 + i]; if (x > nv) { nv = x; ni = s0 + i; } }
            lv[t] = nv; li[t] = ni;
        }
    }
}

// ---------- threefry2x32 (JAX key(42) => k0=0, k1=42) ----------
__device__ __forceinline__ uint32_t rotl32(uint32_t x, int d) { return (x << d) | (x >> (32 - d)); }
__device__ __forceinline__ void threefry2x32(uint32_t k0, uint32_t k1, uint32_t x0, uint32_t x1,
                                             uint32_t* o0, uint32_t* o1) {
    uint32_t ks2 = 0x1BD11BDAu ^ k0 ^ k1;
    x0 += k0; x1 += k1;
#define TF_R(a) { x0 += x1; x1 = rotl32(x1, a); x1 ^= x0; }
    TF_R(13) TF_R(15) TF_R(26) TF_R(6)
    x0 += k1;  x1 += ks2 + 1u;
    TF_R(17) TF_R(29) TF_R(16) TF_R(24)
    x0 += ks2; x1 += k0 + 2u;
    TF_R(13) TF_R(15) TF_R(26) TF_R(6)
    x0 += k0;  x1 += k1 + 3u;
    TF_R(17) TF_R(29) TF_R(16) TF_R(24)
    x0 += k1;  x1 += ks2 + 4u;
#undef TF_R
    *o0 = x0; *o1 = x1;
}

// ---------- K4: top-p/top-k mask + Gumbel-argmax sample ----------
__global__ void __launch_bounds__(64)
k_sample(const float* __restrict__ topv, const int* __restrict__ topi,
         const float* __restrict__ rowMax, const float* __restrict__ rowZ,
         const float* __restrict__ top_ps, const int* __restrict__ top_ks,
         float* __restrict__ out) {
    const int b = blockIdx.x, r = threadIdx.x;   // 64 threads
    __shared__ float p[TOPC]; __shared__ float cex[TOPC];
    __shared__ float sv[TOPC]; __shared__ int  si[TOPC];

    const float v   = topv[b * TOPC + r];
    const int  vidx = topi[b * TOPC + r];
    const float prob = __expf(v - rowMax[b]) / rowZ[b];
    p[r] = prob; __syncthreads();
    if (r == 0) { float c = 0.f; for (int i = 0; i < TOPC; ++i) { cex[i] = c; c += p[i]; } }
    __syncthreads();

    const bool keep = (cex[r] <= top_ps[b]) && (r < top_ks[b]) && (prob > 0.f);
    float score = -__builtin_huge_valf();
    if (keep) {
        const uint32_t N = (uint32_t)B_ * (uint32_t)V_;      // 8,192,000
        const uint32_t halfN = N / 2u;
        uint32_t f = (uint32_t)b * (uint32_t)V_ + (uint32_t)vidx;
        uint32_t c0, c1, o0, o1;
        bool second = (f >= halfN);
        c0 = second ? (f - halfN) : f;
        c1 = second ? f : (f + halfN);
        threefry2x32(0u, 42u, c0, c1, &o0, &o1);
        uint32_t bits = second ? o1 : o0;
        union { uint32_t u; float fl; } cv; cv.u = 0x3f800000u | (bits >> 9);
        float u01 = cv.fl - 1.0f;                             // [0,1)
        const float tiny = 1.1754943508e-38f;
        float u = tiny + u01 * (1.0f - tiny);
        float g = -__logf(-__logf(u));
        score = v + g;   // softmax constants + renorm cancel under argmax
    }
    sv[r] = score; si[r] = vidx; __syncthreads();
    for (int s = 32; s > 0; s >>= 1) {
        if (r < s) {
            if (sv[r + s] > sv[r]) { sv[r] = sv[r + s]; si[r] = si[r + s]; }
        }
        __syncthreads();
    }
    if (r == 0) out[b] = (float)si[0];
}

// ---------- launcher ----------
extern "C" void kernel_launch(void* const* d_in, const int* in_sizes, int n_in,
                              void* d_out, int out_size, void* d_ws, size_t ws_size,
                              hipStream_t stream) {
    const float* emb    = (const float*)d_in[0];
    const float* hs     = (const float*)d_in[1];
    const int*   outpos = (const int*)d_in[2];
    const float* temps  = (const float*)d_in[3];
    const float* top_ps = (const float*)d_in[4];
    const int*   top_ks = (const int*)d_in[5];
    float* out = (float*)d_out;

    float* ws      = (float*)d_ws;
    float* logits  = ws;                                   // B*V floats
    float* rowMax  = ws + (size_t)B_ * V_;                 // B
    float* rowZ    = rowMax + B_;                          // B
    float* topv    = rowZ + B_;                            // B*64
    int*   topi    = (int*)(topv + (size_t)B_ * TOPC);     // B*64

    k_logits_wmma<<<V_ / 16 / 8, 256, 0, stream>>>(emb, hs, outpos, temps, logits);
    k_rowstats<<<B_, 1024, 0, stream>>>(logits, rowMax, rowZ);
    k_top64<<<B_, 1024, 0, stream>>>(logits, topv, topi);
    k_sample<<<B_, TOPC, 0, stream>>>(topv, topi, rowMax, rowZ, top_ps, top_ks, out);
}